// OriginalImplementationPNP_5136780886469
// MI455X (gfx1250) — compile-verified
//
#include <hip/hip_runtime.h>

#define BS      512
#define CLASSES 64
#define GROUP   8
#define D_FEAT  512

typedef __attribute__((ext_vector_type(2))) float v2f;
typedef __attribute__((ext_vector_type(8))) float v8f;

// ---------------------------------------------------------------------------
// Kernel 1: sim = X @ X^T via V_WMMA_F32_16X16X4_F32 (fp32 A/B, fp32 acc).
// One wave per 16x16 output tile; symmetry exploited (by <= bx computed,
// off-diagonal tiles mirrored with vectorized b128 stores).
// A-layout (ISA 7.12.2, 32-bit A 16x4): lane&15 = M row, lane>>4 selects the
// K pair {2h, 2h+1} across the 2 VGPRs; B (4x16) mirrors with N = lane&15.
// C/D: VGPR r -> row r + 8*(lane>>4), col = lane&15.
// ---------------------------------------------------------------------------
__global__ __launch_bounds__(32) void gemm_bbt_wmma(const float* __restrict__ X,
                                                    float* __restrict__ sim) {
  const int bx = blockIdx.x;      // column tile
  const int by = blockIdx.y;      // row tile
  if (by > bx) return;            // symmetric output: upper triangle + diagonal
  const int lane = threadIdx.x & 31;
  const int mr   = lane & 15;
  const int h    = lane >> 4;
  const int row0 = by * 16;
  const int col0 = bx * 16;
  const float* arow = X + (size_t)(row0 + mr) * D_FEAT + 2 * h;
  const float* brow = X + (size_t)(col0 + mr) * D_FEAT + 2 * h;
  v8f c = {};
#pragma unroll 4
  for (int k0 = 0; k0 < D_FEAT; k0 += 4) {
    v2f a = *(const v2f*)(arow + k0);
    v2f b = *(const v2f*)(brow + k0);
    // 8 args: (neg_a, A, neg_b, B, c_mod, C, reuse_a, reuse_b)
    c = __builtin_amdgcn_wmma_f32_16x16x4_f32(false, a, false, b,
                                              (short)0, c, false, false);
  }

  // Direct stores: col fixed per lane, rows strided by BS.
#pragma unroll
  for (int r = 0; r < 8; ++r)
    sim[(size_t)(row0 + r + 8 * h) * BS + (col0 + mr)] = c[r];

  // Mirror stores (uniform branch): for a given lane these are 8 consecutive
  // floats at sim[(col0+mr)*BS + row0 + 8h .. +7], 32B aligned -> 2x b128.
  if (bx != by) {
    float4* p = (float4*)(sim + (size_t)(col0 + mr) * BS + row0 + 8 * h);
    p[0] = make_float4(c[0], c[1], c[2], c[3]);
    p[1] = make_float4(c[4], c[5], c[6], c[7]);
  }
}

// ---------------------------------------------------------------------------
// Kernel 2: diagonal-block rank sums.
// Block = one class c (64 blocks), 8 waves, wave w handles row i = c*8 + w.
// Each lane caches 16 strided sim[i,k] values + the 0/1 mask in registers,
// reused across the 8 in-block columns j. mask[j,k] for j in block c reduces
// to (k>>3 != c) (the -eye diagonal lies inside the class block).
// expo clamp to [-50,50] matches the reference jnp.clip before exp, and also
// guarantees exp() is finite so the raw v_rcp_f32 is safe (~1 ulp per term
// inside a 512-term sum).
// ---------------------------------------------------------------------------
__global__ __launch_bounds__(256) void rank_kernel(const float* __restrict__ sim,
                                                   float* __restrict__ acc) {
  const int c    = blockIdx.x;            // class 0..63
  const int wave = threadIdx.x >> 5;      // 0..7
  const int lane = threadIdx.x & 31;
  const int i    = c * GROUP + wave;      // global row
  const float* srow = sim + (size_t)i * BS;

  float simk[16];
  float maskf[16];
#pragma unroll
  for (int t = 0; t < 16; ++t) {
    const int k = lane + 32 * t;
    simk[t]  = srow[k];
    maskf[t] = ((k >> 3) == c) ? 0.0f : 1.0f;
  }

  float waveacc = 0.0f;
#pragma unroll
  for (int jj = 0; jj < GROUP; ++jj) {
    const float sij = srow[c * GROUP + jj];   // uniform across the wave
    float s = 0.0f;
#pragma unroll
    for (int t = 0; t < 16; ++t) {
      float e = (sij - simk[t]) * 100.0f;     // = clip(-sim_diff/ANNEAL)
      e = fminf(fmaxf(e, -50.0f), 50.0f);
      s += maskf[t] * __builtin_amdgcn_rcpf(1.0f + __expf(e));  // v_rcp_f32
    }
    // wave32 butterfly reduction
#pragma unroll
    for (int off = 16; off > 0; off >>= 1)
      s += __shfl_xor(s, off, 32);
    const float r  = 1.0f / (1.0f + s);       // 1/(1+rk)^ALPHA, ALPHA=4
    const float r2 = r * r;
    waveacc += r2 * r2;
  }
  if (lane == 0) atomicAdd(acc, waveacc);
}

__global__ void zero_acc(float* __restrict__ acc) { *acc = 0.0f; }

__global__ void finalize_k(const float* __restrict__ acc, float* __restrict__ out) {
  out[0] = 1.0f - acc[0] * (1.0f / (float)(GROUP * BS));   // /4096
}

// ---------------------------------------------------------------------------
extern "C" void kernel_launch(void* const* d_in, const int* in_sizes, int n_in,
                              void* d_out, int out_size, void* d_ws, size_t ws_size,
                              hipStream_t stream) {
  (void)in_sizes; (void)n_in; (void)out_size; (void)ws_size;
  const float* batch = (const float*)d_in[0];
  float* sim = (float*)d_ws;                       // 512*512 floats = 1 MB
  float* acc = sim + (size_t)BS * BS;              // +4 bytes scratch
  float* out = (float*)d_out;

  zero_acc<<<dim3(1), dim3(1), 0, stream>>>(acc);
  gemm_bbt_wmma<<<dim3(BS / 16, BS / 16), dim3(32), 0, stream>>>(batch, sim);
  rank_kernel<<<dim3(CLASSES), dim3(256), 0, stream>>>(sim, acc);
  finalize_k<<<dim3(1), dim3(1), 0, stream>>>(acc, out);
}